// GraphGRUCell_62019327754706
// MI455X (gfx1250) — compile-verified
//
#include <hip/hip_runtime.h>
#include <hip/hip_bf16.h>
#include <math.h>

typedef __attribute__((ext_vector_type(2))) float v2f;
typedef __attribute__((ext_vector_type(8))) float v8f;

#define Bdim 4
#define Ndim 20000
#define Udim 64
#define Edim 640000
#define Fdim 65        // U + D
#define KPAD 68        // F padded to multiple of 4 for 16x16x4 WMMA K-loop
#define NEGV (-1e30f)
#define SCALEF 0.125f  // 1/sqrt(64)

// ---------------------------------------------------------------------------
// K0: init workspace accumulators (harness poisons ws with 0xAA)
// ---------------------------------------------------------------------------
__global__ __launch_bounds__(256) void k0_init(float* __restrict__ NUM,
                                               float* __restrict__ MX,
                                               float* __restrict__ DEN) {
  int idx = blockIdx.x * 256 + threadIdx.x;
  if (idx < Bdim * Ndim * Udim) NUM[idx] = 0.f;
  if (idx < Bdim * Ndim) { MX[idx] = NEGV; DEN[idx] = 0.f; }
}

// ---------------------------------------------------------------------------
// K1: q,k,v,s = x @ {Wq,Wk,Wv,Ws} + bias    (x = [state | inputs], 80000x65)
// One block = 16 rows; 8 waves x 2 tiles = 16 (4 matrices x 4 col-tiles of 16)
// WMMA f32 16x16x4, fully unrolled K-loop over 68 (padded)
// ---------------------------------------------------------------------------
__global__ __launch_bounds__(256) void k1_qkvs(
    const float* __restrict__ inputs, const float* __restrict__ state,
    const float* __restrict__ Wq, const float* __restrict__ bq,
    const float* __restrict__ Wk, const float* __restrict__ bk,
    const float* __restrict__ Wv, const float* __restrict__ bv,
    const float* __restrict__ Ws, const float* __restrict__ bs,
    float* __restrict__ Q, float* __restrict__ Kb,
    float* __restrict__ V, float* __restrict__ S) {
  __shared__ float xs[16][KPAD];
  const int rowbase = blockIdx.x * 16;
  for (int t = threadIdx.x; t < 16 * KPAD; t += 256) {
    int r = t / KPAD, c = t - r * KPAD;
    int row = rowbase + r;
    float v = 0.f;
    if (c < Udim) v = state[row * Udim + c];
    else if (c == Udim) v = inputs[row];
    xs[r][c] = v;
  }
  __syncthreads();
  const int wave = threadIdx.x >> 5;
  const int lane = threadIdx.x & 31;
  const int l15  = lane & 15;   // A-row (M) and B-col (N) for this lane
  const int kh   = lane >> 4;   // K-half select
  for (int t2 = 0; t2 < 2; ++t2) {
    const int tile = wave * 2 + t2;            // 0..15
    const int mat  = tile >> 2;                // which weight matrix
    const int cb   = (tile & 3) * 16;          // column base within U=64
    const float* W  = (mat == 0) ? Wq : (mat == 1) ? Wk : (mat == 2) ? Wv : Ws;
    const float* bb = (mat == 0) ? bq : (mat == 1) ? bk : (mat == 2) ? bv : bs;
    float*       O  = (mat == 0) ? Q  : (mat == 1) ? Kb : (mat == 2) ? V  : S;
    // Preload B fragments (W is block-invariant; batches into load clauses)
    float bx[KPAD / 4], by[KPAD / 4];
#pragma unroll
    for (int i = 0; i < KPAD / 4; ++i) {
      const int ka = 4 * i + 2 * kh;
      bx[i] = (ka     < Fdim) ? W[ka * Udim + cb + l15]       : 0.f;
      by[i] = (ka + 1 < Fdim) ? W[(ka + 1) * Udim + cb + l15] : 0.f;
    }
    v8f acc = {};
#pragma unroll
    for (int i = 0; i < KPAD / 4; ++i) {
      const int ka = 4 * i + 2 * kh;
      v2f a, b;
      a.x = xs[l15][ka];
      a.y = xs[l15][ka + 1];
      b.x = bx[i];
      b.y = by[i];
      acc = __builtin_amdgcn_wmma_f32_16x16x4_f32(false, a, false, b,
                                                  (short)0, acc, false, false);
    }
    const float bias = bb[cb + l15];
#pragma unroll
    for (int r = 0; r < 8; ++r) {
      const int mm = r + 8 * kh;  // C/D layout: lanes 16-31 hold M=8..15
      O[(rowbase + mm) * Udim + cb + l15] = acc[r] + bias;
    }
  }
}

// ---------------------------------------------------------------------------
// K2: per-edge masked score + segment max.  16 lanes per (b,edge), 2/wave.
// ---------------------------------------------------------------------------
__global__ __launch_bounds__(256) void k2_scores(
    const int* __restrict__ esrc, const int* __restrict__ edst,
    const float* __restrict__ state,
    const float* __restrict__ Q, const float* __restrict__ Kb,
    float* __restrict__ SC, float* __restrict__ MX) {
  const int grp = blockIdx.x * 16 + (threadIdx.x >> 4);  // edge-instance id
  const int l   = threadIdx.x & 15;
  const int b = grp / Edim;
  const int e = grp - b * Edim;
  const int s = esrc[e];
  const int d = edst[e];
  const int nd = b * Ndim + d;
  const int ns = b * Ndim + s;
  const float4 qv = *(const float4*)(Q  + nd * Udim + 4 * l);
  const float4 kv = *(const float4*)(Kb + ns * Udim + 4 * l);
  float p = qv.x * kv.x + qv.y * kv.y + qv.z * kv.z + qv.w * kv.w;
#pragma unroll
  for (int off = 8; off > 0; off >>= 1) p += __shfl_xor(p, off, 16);
  if (l == 0) {
    const bool md = state[nd * Udim + 58] != 0.f;  // node_mask = x[..,-7]!=0
    const bool ms = state[ns * Udim + 58] != 0.f;
    const float score = (md && ms) ? p * SCALEF : NEGV;
    SC[b * Edim + e] = score;
    if (score > -1e29f) unsafeAtomicMax(&MX[nd], score);  // global_atomic_max_num_f32
  }
}

// ---------------------------------------------------------------------------
// K3: per-edge exp + scatter-add of e*v into num, e into denom.
// ---------------------------------------------------------------------------
__global__ __launch_bounds__(256) void k3_scatter(
    const int* __restrict__ esrc, const int* __restrict__ edst,
    const float* __restrict__ SC, const float* __restrict__ MX,
    const float* __restrict__ V,
    float* __restrict__ NUM, float* __restrict__ DEN) {
  const int grp = blockIdx.x * 16 + (threadIdx.x >> 4);
  const int l   = threadIdx.x & 15;
  const int b = grp / Edim;
  const int e = grp - b * Edim;
  const float sc = SC[b * Edim + e];
  if (sc <= -1e29f) return;  // masked edge
  const int s = esrc[e];
  const int d = edst[e];
  const int nd = b * Ndim + d;
  const int ns = b * Ndim + s;
  const float ev = __expf(sc - MX[nd]);
  const float4 vv = *(const float4*)(V + ns * Udim + 4 * l);
  float* np = &NUM[nd * Udim + 4 * l];
  unsafeAtomicAdd(np + 0, ev * vv.x);   // global_atomic_add_f32
  unsafeAtomicAdd(np + 1, ev * vv.y);
  unsafeAtomicAdd(np + 2, ev * vv.z);
  unsafeAtomicAdd(np + 3, ev * vv.w);
  if (l == 0) unsafeAtomicAdd(&DEN[nd], ev);
}

// ---------------------------------------------------------------------------
// K4: h2 = mask ? (num/denom + s) : h
// ---------------------------------------------------------------------------
__global__ __launch_bounds__(256) void k4_h2(
    const float* __restrict__ state, const float* __restrict__ S,
    const float* __restrict__ NUM, const float* __restrict__ DEN,
    float* __restrict__ H2) {
  const int idx = blockIdx.x * 256 + threadIdx.x;
  if (idx >= Bdim * Ndim * Udim) return;
  const int n = idx >> 6;
  const float den = DEN[n];
  const float agg = (den > 0.f) ? NUM[idx] / fmaxf(den, 1e-16f) : 0.f;
  const float out = agg + S[idx];
  const bool mask = state[n * Udim + 58] != 0.f;
  H2[idx] = mask ? out : state[idx];
}

// ---------------------------------------------------------------------------
// K5: value = sigmoid([xin | h2] @ W1 + b1); emit rh = reset*h2 and z.
// 8 waves -> 8 column tiles of 16 (128 outputs).
// ---------------------------------------------------------------------------
__global__ __launch_bounds__(256) void k5_gate(
    const float* __restrict__ inputs, const float* __restrict__ H2,
    const float* __restrict__ W1, const float* __restrict__ b1,
    float* __restrict__ RH, float* __restrict__ Z) {
  __shared__ float xs[16][KPAD];
  const int rowbase = blockIdx.x * 16;
  for (int t = threadIdx.x; t < 16 * KPAD; t += 256) {
    int r = t / KPAD, c = t - r * KPAD;
    int row = rowbase + r;
    float v = 0.f;
    if (c == 0) v = inputs[row];
    else if (c <= Udim) v = H2[row * Udim + (c - 1)];
    xs[r][c] = v;
  }
  __syncthreads();
  const int wave = threadIdx.x >> 5;
  const int lane = threadIdx.x & 31;
  const int l15  = lane & 15;
  const int kh   = lane >> 4;
  const int cb   = wave * 16;  // output col base within 128
  float bx[KPAD / 4], by[KPAD / 4];
#pragma unroll
  for (int i = 0; i < KPAD / 4; ++i) {
    const int ka = 4 * i + 2 * kh;
    bx[i] = (ka     < Fdim) ? W1[ka * 128 + cb + l15]       : 0.f;
    by[i] = (ka + 1 < Fdim) ? W1[(ka + 1) * 128 + cb + l15] : 0.f;
  }
  v8f acc = {};
#pragma unroll
  for (int i = 0; i < KPAD / 4; ++i) {
    const int ka = 4 * i + 2 * kh;
    v2f a, b;
    a.x = xs[l15][ka];
    a.y = xs[l15][ka + 1];
    b.x = bx[i];
    b.y = by[i];
    acc = __builtin_amdgcn_wmma_f32_16x16x4_f32(false, a, false, b,
                                                (short)0, acc, false, false);
  }
  const int j = cb + l15;
  const float bias = b1[j];
#pragma unroll
  for (int r = 0; r < 8; ++r) {
    const int row = rowbase + r + 8 * kh;
    const float sg = 1.f / (1.f + __expf(-(acc[r] + bias)));
    if (j < Udim) RH[row * Udim + j] = sg * H2[row * Udim + j];   // reset*h2
    else          Z[row * Udim + (j - Udim)] = sg;                // z
  }
}

// ---------------------------------------------------------------------------
// K6: c = tanh([xin | rh] @ W2 + b2); new_state = (1-z)*h2 + z*c.
// 4 waves -> 4 column tiles of 16 (64 outputs).
// ---------------------------------------------------------------------------
__global__ __launch_bounds__(128) void k6_out(
    const float* __restrict__ inputs, const float* __restrict__ RH,
    const float* __restrict__ H2, const float* __restrict__ Z,
    const float* __restrict__ W2, const float* __restrict__ b2,
    float* __restrict__ out) {
  __shared__ float xs[16][KPAD];
  const int rowbase = blockIdx.x * 16;
  for (int t = threadIdx.x; t < 16 * KPAD; t += 128) {
    int r = t / KPAD, c = t - r * KPAD;
    int row = rowbase + r;
    float v = 0.f;
    if (c == 0) v = inputs[row];
    else if (c <= Udim) v = RH[row * Udim + (c - 1)];
    xs[r][c] = v;
  }
  __syncthreads();
  const int wave = threadIdx.x >> 5;
  const int lane = threadIdx.x & 31;
  const int l15  = lane & 15;
  const int kh   = lane >> 4;
  const int cb   = wave * 16;
  float bx[KPAD / 4], by[KPAD / 4];
#pragma unroll
  for (int i = 0; i < KPAD / 4; ++i) {
    const int ka = 4 * i + 2 * kh;
    bx[i] = (ka     < Fdim) ? W2[ka * Udim + cb + l15]       : 0.f;
    by[i] = (ka + 1 < Fdim) ? W2[(ka + 1) * Udim + cb + l15] : 0.f;
  }
  v8f acc = {};
#pragma unroll
  for (int i = 0; i < KPAD / 4; ++i) {
    const int ka = 4 * i + 2 * kh;
    v2f a, b;
    a.x = xs[l15][ka];
    a.y = xs[l15][ka + 1];
    b.x = bx[i];
    b.y = by[i];
    acc = __builtin_amdgcn_wmma_f32_16x16x4_f32(false, a, false, b,
                                                (short)0, acc, false, false);
  }
  const int j = cb + l15;
  const float bias = b2[j];
#pragma unroll
  for (int r = 0; r < 8; ++r) {
    const int row = rowbase + r + 8 * kh;
    const float c  = tanhf(acc[r] + bias);
    const float h2 = H2[row * Udim + j];
    const float z  = Z[row * Udim + j];
    out[row * Udim + j] = (1.f - z) * h2 + z * c;
  }
}

// ---------------------------------------------------------------------------
extern "C" void kernel_launch(void* const* d_in, const int* in_sizes, int n_in,
                              void* d_out, int out_size, void* d_ws, size_t ws_size,
                              hipStream_t stream) {
  const float* inputs = (const float*)d_in[0];
  const float* state  = (const float*)d_in[1];
  const int*   esrc   = (const int*)d_in[2];
  const int*   edst   = (const int*)d_in[3];
  const float* Wq = (const float*)d_in[4];  const float* bq = (const float*)d_in[5];
  const float* Wk = (const float*)d_in[6];  const float* bk = (const float*)d_in[7];
  const float* Wv = (const float*)d_in[8];  const float* bv = (const float*)d_in[9];
  const float* Ws = (const float*)d_in[10]; const float* bs = (const float*)d_in[11];
  const float* W1 = (const float*)d_in[12]; const float* b1 = (const float*)d_in[13];
  const float* W2 = (const float*)d_in[14]; const float* b2 = (const float*)d_in[15];
  float* out = (float*)d_out;

  // workspace layout (floats). RH aliases Q (dead after K2), Z aliases K.
  const size_t NU = (size_t)Bdim * Ndim * Udim;   // 5,120,000
  const size_t BN = (size_t)Bdim * Ndim;          // 80,000
  const size_t BE = (size_t)Bdim * Edim;          // 2,560,000
  float* ws = (float*)d_ws;
  float* Q   = ws;             float* RH = Q;  // alias
  float* Kb  = ws + NU;        float* Z  = Kb; // alias
  float* V   = ws + 2 * NU;
  float* S   = ws + 3 * NU;
  float* NUM = ws + 4 * NU;
  float* H2  = ws + 5 * NU;
  float* SC  = ws + 6 * NU;
  float* MX  = ws + 6 * NU + BE;
  float* DEN = ws + 6 * NU + BE + BN;

  const int rowTiles   = (Bdim * Ndim) / 16;            // 5000
  const int elemBlocks = (int)((NU + 255) / 256);       // 20000
  const int edgeBlocks = (int)(BE / 16);                // 160000 (16 lanes/edge)

  k0_init<<<elemBlocks, 256, 0, stream>>>(NUM, MX, DEN);
  k1_qkvs<<<rowTiles, 256, 0, stream>>>(inputs, state, Wq, bq, Wk, bk, Wv, bv,
                                        Ws, bs, Q, Kb, V, S);
  k2_scores<<<edgeBlocks, 256, 0, stream>>>(esrc, edst, state, Q, Kb, SC, MX);
  k3_scatter<<<edgeBlocks, 256, 0, stream>>>(esrc, edst, SC, MX, V, NUM, DEN);
  k4_h2<<<elemBlocks, 256, 0, stream>>>(state, S, NUM, DEN, H2);
  k5_gate<<<rowTiles, 256, 0, stream>>>(inputs, H2, W1, b1, RH, Z);
  k6_out<<<rowTiles, 128, 0, stream>>>(inputs, RH, H2, Z, W2, b2, out);
}